// NERModel_37297495998611
// MI455X (gfx1250) — compile-verified
//
#include <hip/hip_runtime.h>
#include <hip/hip_bf16.h>
#include <math.h>

// ---------------------------------------------------------------------------
// Problem dims (fixed by reference)
// ---------------------------------------------------------------------------
#define NUM_TAGS 9
#define BB 64
#define SS 512
#define HH 768
#define ROWS (BB * SS)          // 32768
#define EM_STRIDE 16            // padded tag dimension
#define LDS_W_STRIDE 772        // 768 + 4 pad: even (8B aligned b64) + bank spread

typedef __attribute__((ext_vector_type(2))) float v2f;
typedef __attribute__((ext_vector_type(8))) float v8f;

// ---------------------------------------------------------------------------
// Kernel 1: emissions[r, k] = sum_h hidden[r,h] * W[k,h] + b[k]
// One wave (32 lanes) computes a 16(row) x 16(tag) tile with
// V_WMMA_F32_16X16X4_F32, K-loop over 768 in steps of 4.
//
// fp32 A 16x4 layout : lanes 0-15 -> M=lane,  VGPR0=K0 VGPR1=K1
//                      lanes 16-31-> M=lane-16, VGPR0=K2 VGPR1=K3
// fp32 B 4x16 layout : lanes 0-15 -> N=lane,  VGPR0=K0 VGPR1=K1
//                      lanes 16-31-> N=lane-16, VGPR0=K2 VGPR1=K3
// fp32 C/D 16x16     : VGPR r: lanes 0-15 -> M=r, lanes 16-31 -> M=8+r; N=lane&15
// ---------------------------------------------------------------------------
__global__ __launch_bounds__(128)
void ner_emissions_wmma(const float* __restrict__ hidden,
                        const float* __restrict__ W,
                        const float* __restrict__ bias,
                        float* __restrict__ em) {
    __shared__ float lw[16 * LDS_W_STRIDE];   // W zero-padded to 16 tags

    const int tid = threadIdx.x;
    // Stage W into LDS (tags 9..15 zero)
    for (int idx = tid; idx < 16 * HH; idx += 128) {
        const int tag = idx / HH;
        const int h   = idx - tag * HH;
        lw[tag * LDS_W_STRIDE + h] = (tag < NUM_TAGS) ? W[tag * HH + h] : 0.0f;
    }
    __syncthreads();

    const int wave     = tid >> 5;                       // 4 waves / block
    const int lane     = tid & 31;
    const int tileIdx  = blockIdx.x * 4 + wave;          // grid sized exactly
    const int row0     = tileIdx * 16;

    const int lr   = lane & 15;            // A: row-in-tile; B: tag (column)
    const int koff = (lane >> 4) * 2;      // K offset within the 4-wide step

    const float* hrow = hidden + (size_t)(row0 + lr) * HH + koff;
    const float* wrow = &lw[lr * LDS_W_STRIDE + koff];

    v8f acc = {};
    #pragma unroll 4
    for (int k = 0; k < HH; k += 4) {
        v2f a = *(const v2f*)(hrow + k);   // global_load_b64
        v2f w = *(const v2f*)(wrow + k);   // ds_load_b64 (8B aligned, stride 772)
        acc = __builtin_amdgcn_wmma_f32_16x16x4_f32(
            /*neg_a=*/false, a, /*neg_b=*/false, w,
            /*c_mod=*/(short)0, acc, /*reuse_a=*/false, /*reuse_b=*/false);
    }

    // bias (zero for padded tags), then store D tile: coalesced 64B per half-wave
    const int   ccol = lane & 15;
    const float bv   = (ccol < NUM_TAGS) ? bias[ccol] : 0.0f;
    const int   crow = row0 + (lane >> 4) * 8;
    #pragma unroll
    for (int r = 0; r < 8; ++r) {
        em[(size_t)(crow + r) * EM_STRIDE + ccol] = acc[r] + bv;
    }
}

// ---------------------------------------------------------------------------
// Kernel 2: per-batch gold score + CRF forward scan (one wave per batch).
// Lane j (<9) owns alpha[j]; cross-lane reads via __shfl (wave32).
// ---------------------------------------------------------------------------
__global__ __launch_bounds__(32)
void ner_crf_scan(const float* __restrict__ em,
                  const float* __restrict__ start_trans,
                  const float* __restrict__ end_trans,
                  const float* __restrict__ trans,
                  const int*   __restrict__ labels,
                  const int*   __restrict__ mask,
                  float* __restrict__ llh_out) {
    const int b    = blockIdx.x;
    const int lane = threadIdx.x;

    const float* emB = em + (size_t)b * SS * EM_STRIDE;
    const int*   lab = labels + (size_t)b * SS;
    const int*   mk  = mask   + (size_t)b * SS;

    // ---- gold score partials (strided over t, reduced across wave) ----
    float part = 0.0f;
    int   mcnt = 0;
    for (int t = lane; t < SS; t += 32) {
        const int m = mk[t];
        mcnt += m;
        if (t >= 1) {
            const int lt  = lab[t];
            const int ltp = lab[t - 1];
            const float c = emB[t * EM_STRIDE + lt] + trans[ltp * NUM_TAGS + lt];
            part += (m > 0) ? c : 0.0f;
        }
    }
    #pragma unroll
    for (int off = 16; off >= 1; off >>= 1) {
        part += __shfl_xor(part, off, 32);
        mcnt += __shfl_xor(mcnt, off, 32);
    }

    // ---- forward scan ----
    const bool active = (lane < NUM_TAGS);
    float tcol[NUM_TAGS];
    #pragma unroll
    for (int i = 0; i < NUM_TAGS; ++i)
        tcol[i] = active ? trans[i * NUM_TAGS + lane] : 0.0f;

    float alpha = active ? (start_trans[lane] + emB[lane]) : -1e30f;

    for (int t = 1; t < SS; ++t) {
        const float emj = active ? emB[t * EM_STRIDE + lane] : 0.0f;
        const bool  upd = (mk[t] > 0);

        float av[NUM_TAGS];
        float mx = -1e30f;
        #pragma unroll
        for (int i = 0; i < NUM_TAGS; ++i) {
            const float ai = __shfl(alpha, i, 32);   // all lanes participate
            av[i] = ai + tcol[i];
            mx = fmaxf(mx, av[i]);
        }
        float s = 0.0f;
        #pragma unroll
        for (int i = 0; i < NUM_TAGS; ++i)
            s += __expf(av[i] - mx);
        const float nxt = mx + __logf(s) + emj;
        if (upd && active) alpha = nxt;
    }

    // ---- denom = logsumexp_j(alpha[j] + end_trans[j]) ----
    float xv = active ? (alpha + end_trans[lane]) : -1e30f;
    float mx = xv;
    #pragma unroll
    for (int off = 16; off >= 1; off >>= 1)
        mx = fmaxf(mx, __shfl_xor(mx, off, 32));
    float s = active ? __expf(xv - mx) : 0.0f;
    #pragma unroll
    for (int off = 16; off >= 1; off >>= 1)
        s += __shfl_xor(s, off, 32);
    const float denom = mx + __logf(s);

    if (lane == 0) {
        const int   l0    = lab[0];
        float score = start_trans[l0] + emB[l0] + part;
        const int   send  = mcnt - 1;              // seq_ends
        score += end_trans[lab[send]];
        llh_out[b] = score - denom;
    }
}

// ---------------------------------------------------------------------------
// Kernel 3: out = -mean(llh) over 64 batches, deterministic wave reduction.
// ---------------------------------------------------------------------------
__global__ __launch_bounds__(32)
void ner_final_reduce(const float* __restrict__ llh, float* __restrict__ out) {
    const int lane = threadIdx.x;
    float v = llh[lane] + llh[lane + 32];
    #pragma unroll
    for (int off = 16; off >= 1; off >>= 1)
        v += __shfl_xor(v, off, 32);
    if (lane == 0) out[0] = -(v / (float)BB);
}

// ---------------------------------------------------------------------------
extern "C" void kernel_launch(void* const* d_in, const int* in_sizes, int n_in,
                              void* d_out, int out_size, void* d_ws, size_t ws_size,
                              hipStream_t stream) {
    const float* hidden      = (const float*)d_in[0];
    const float* W           = (const float*)d_in[1];
    const float* bias        = (const float*)d_in[2];
    const float* start_trans = (const float*)d_in[3];
    const float* end_trans   = (const float*)d_in[4];
    const float* trans       = (const float*)d_in[5];
    const int*   labels      = (const int*)d_in[6];
    const int*   mask        = (const int*)d_in[7];
    float*       out         = (float*)d_out;

    float* em  = (float*)d_ws;                          // 32768*16 floats = 2 MB
    float* llh = em + (size_t)ROWS * EM_STRIDE;         // 64 floats

    // K1: 2048 tiles of 16 rows, 4 waves per block -> 512 blocks x 128 threads
    ner_emissions_wmma<<<ROWS / 16 / 4, 128, 0, stream>>>(hidden, W, bias, em);
    // K2: one wave per batch
    ner_crf_scan<<<BB, 32, 0, stream>>>(em, start_trans, end_trans, trans,
                                        labels, mask, llh);
    // K3: final scalar
    ner_final_reduce<<<1, 32, 0, stream>>>(llh, out);
}